// NGCFModel_6811818132464
// MI455X (gfx1250) — compile-verified
//
#include <hip/hip_runtime.h>
#include <hip/hip_bf16.h>

#define N_USERS 100000
#define N_ITEMS 50000
#define NTOT    (N_USERS + N_ITEMS)   // 150000
#define EMBED   64
#define OUTD    256                   // 64 ego + 3*64 layers
#define NEG_SLOPE 0.2f
#define L2EPS   1e-12f
#define LDS_STRIDE 66                 // even (8B-aligned b64 reads), low bank conflict

typedef float v2f __attribute__((ext_vector_type(2)));
typedef float v8f __attribute__((ext_vector_type(8)));

// ---------------------------------------------------------------------------
// ego = concat(user_embed, item_embed) -> out[:, 0:64]   (float4 vectorized)
// ---------------------------------------------------------------------------
__global__ void k_init(const float* __restrict__ ue, const float* __restrict__ ie,
                       float* __restrict__ out) {
    long t = (long)blockIdx.x * blockDim.x + threadIdx.x;   // one float4 each
    const long total = (long)NTOT * (EMBED / 4);
    if (t >= total) return;
    int row = (int)(t / (EMBED / 4));
    int c4  = (int)(t % (EMBED / 4));
    float4 v;
    if (row < N_USERS)
        v = ((const float4*)ue)[(long)row * (EMBED / 4) + c4];
    else
        v = ((const float4*)ie)[(long)(row - N_USERS) * (EMBED / 4) + c4];
    ((float4*)(out + (long)row * OUTD))[c4] = v;
}

// ---------------------------------------------------------------------------
// zero the side accumulator (N x 64 floats), float4 vectorized
// ---------------------------------------------------------------------------
__global__ void k_zero(float* __restrict__ p, long n4) {
    long t = (long)blockIdx.x * blockDim.x + threadIdx.x;
    if (t >= n4) return;
    ((float4*)p)[t] = make_float4(0.f, 0.f, 0.f, 0.f);
}

// ---------------------------------------------------------------------------
// SpMM: side[rows[e], :] += vals[e] * prev[cols[e], :]
// prev lives in out[:, colbase:colbase+64].  32 threads per edge, float2/lane.
// ---------------------------------------------------------------------------
__global__ void k_spmm(const int* __restrict__ rows, const int* __restrict__ cols,
                       const float* __restrict__ vals, const float* __restrict__ out,
                       int colbase, float* __restrict__ side, int nnz) {
    long gid = (long)blockIdx.x * blockDim.x + threadIdx.x;
    int e = (int)(gid >> 5);
    if (e >= nnz) return;
    int l = (int)(gid & 31);
    int r = rows[e];
    int c = cols[e];
    float v = vals[e];
    float2 p = *(const float2*)(out + (long)c * OUTD + colbase + 2 * l);
    float* d = side + (long)r * EMBED + 2 * l;
    unsafeAtomicAdd(d,     v * p.x);   // global_atomic_add_f32
    unsafeAtomicAdd(d + 1, v * p.y);
}

// ---------------------------------------------------------------------------
// Dense stage per layer, WMMA f32 16x16x4:
//   msg = side@Ws + (side*prev)@Wp + (bs+bp);  leaky_relu;  row-L2-normalize
//   prev = out[:, cin:cin+64]; result -> out[:, cout:cout+64]
// One wave per 16-row tile; 8 waves (256 threads) per block.
// ---------------------------------------------------------------------------
__global__ __launch_bounds__(256) void k_dense(
    const float* __restrict__ side, const float* __restrict__ out_in,
    float* __restrict__ out,
    const float* __restrict__ Ws, const float* __restrict__ bs,
    const float* __restrict__ Wp, const float* __restrict__ bp,
    int cin, int cout)
{
    __shared__ float WsT[64 * LDS_STRIDE];   // WsT[n][k] = Ws[k][n]
    __shared__ float WpT[64 * LDS_STRIDE];
    __shared__ float bsum[64];

    const int tid = threadIdx.x;
    for (int idx = tid; idx < 64 * 64; idx += 256) {
        int k = idx >> 6, n = idx & 63;
        WsT[n * LDS_STRIDE + k] = Ws[idx];
        WpT[n * LDS_STRIDE + k] = Wp[idx];
    }
    if (tid < 64) bsum[tid] = bs[tid] + bp[tid];
    __syncthreads();

    const int wave = tid >> 5;
    const int lane = tid & 31;
    const int tile = blockIdx.x * 8 + wave;
    if (tile >= NTOT / 16) return;           // whole wave retires together

    const int row0 = tile * 16;
    const int m  = lane & 15;                // A-row / B-col index for this lane
    const int hi = lane >> 4;                // half-wave select (K group)
    const int row = row0 + m;

    // --- preload A fragments: a1 = side tile, a2 = side*prev tile -----------
    // 16x4 f32 A layout: lane = m + 16*(k>>1) ... element (m, ks*4 + 2*hi + j)
    v2f a1[16], a2[16];
    const float* srow = side   + (long)row * EMBED;
    const float* prow = out_in + (long)row * OUTD + cin;
#pragma unroll
    for (int ks = 0; ks < 16; ++ks) {
        int k0 = ks * 4 + 2 * hi;
        float2 s2 = *(const float2*)(srow + k0);
        float2 p2 = *(const float2*)(prow + k0);
        v2f a; a.x = s2.x;        a.y = s2.y;        a1[ks] = a;
        v2f b; b.x = s2.x * p2.x; b.y = s2.y * p2.y; a2[ks] = b;
    }

    // --- 4 output column tiles, K=64 in 16 steps of 2 WMMAs each ------------
    v8f msg[4];
#pragma unroll
    for (int nt = 0; nt < 4; ++nt) {
        v8f acc = {};
        const float* wsrow = &WsT[(nt * 16 + m) * LDS_STRIDE];
        const float* wprow = &WpT[(nt * 16 + m) * LDS_STRIDE];
#pragma unroll
        for (int ks = 0; ks < 16; ++ks) {
            int k0 = ks * 4 + 2 * hi;
            float2 f2 = *(const float2*)(wsrow + k0);   // ds_load_b64
            v2f bsv; bsv.x = f2.x; bsv.y = f2.y;
            acc = __builtin_amdgcn_wmma_f32_16x16x4_f32(
                      false, a1[ks], false, bsv, (short)0, acc, false, false);
            float2 g2 = *(const float2*)(wprow + k0);
            v2f bpv; bpv.x = g2.x; bpv.y = g2.y;
            acc = __builtin_amdgcn_wmma_f32_16x16x4_f32(
                      false, a2[ks], false, bpv, (short)0, acc, false, false);
        }
        msg[nt] = acc;
    }

    // --- bias + leaky relu + per-row sum of squares -------------------------
    // C layout: lane holds (M = vg + 8*hi, N = nt*16 + (lane&15))
    float ss[8];
#pragma unroll
    for (int vg = 0; vg < 8; ++vg) ss[vg] = 0.f;
#pragma unroll
    for (int nt = 0; nt < 4; ++nt) {
        float bb = bsum[nt * 16 + m];
#pragma unroll
        for (int vg = 0; vg < 8; ++vg) {
            float v = msg[nt][vg] + bb;
            v = (v >= 0.f) ? v : NEG_SLOPE * v;
            msg[nt][vg] = v;
            ss[vg] += v * v;
        }
    }

    // reduce across the 16 lanes of each half-wave (masks 1..8 never cross bit4)
#pragma unroll
    for (int vg = 0; vg < 8; ++vg) {
        float s = ss[vg];
        s += __shfl_xor(s, 1, 32);
        s += __shfl_xor(s, 2, 32);
        s += __shfl_xor(s, 4, 32);
        s += __shfl_xor(s, 8, 32);
        float nrm = fmaxf(sqrtf(s), L2EPS);
        ss[vg] = 1.f / nrm;
    }

    // --- scaled store into out[:, cout:cout+64] -----------------------------
#pragma unroll
    for (int vg = 0; vg < 8; ++vg) {
        int mout = vg + 8 * hi;
        float* orow = out + (long)(row0 + mout) * OUTD + cout;
#pragma unroll
        for (int nt = 0; nt < 4; ++nt)
            orow[nt * 16 + m] = msg[nt][vg] * ss[vg];
    }
}

// ---------------------------------------------------------------------------
extern "C" void kernel_launch(void* const* d_in, const int* in_sizes, int n_in,
                              void* d_out, int out_size, void* d_ws, size_t ws_size,
                              hipStream_t stream) {
    const int*   rows = (const int*)d_in[0];
    const int*   cols = (const int*)d_in[1];
    const float* vals = (const float*)d_in[2];
    const float* ue   = (const float*)d_in[3];
    const float* ie   = (const float*)d_in[4];
    const float* W[3][4];
    for (int k = 0; k < 3; ++k)
        for (int j = 0; j < 4; ++j)
            W[k][j] = (const float*)d_in[5 + 4 * k + j];

    float* out  = (float*)d_out;
    float* side = (float*)d_ws;          // NTOT x 64 floats = 38.4 MB scratch
    const int nnz = in_sizes[0];

    {   // ego -> out[:, 0:64]
        long total = (long)NTOT * (EMBED / 4);
        unsigned blocks = (unsigned)((total + 255) / 256);
        k_init<<<blocks, 256, 0, stream>>>(ue, ie, out);
    }

    const long zero4       = (long)NTOT * EMBED / 4;
    const unsigned zblocks = (unsigned)((zero4 + 255) / 256);
    const long sthreads    = (long)nnz * 32;
    const unsigned sblocks = (unsigned)((sthreads + 255) / 256);
    const unsigned dblocks = (unsigned)((NTOT / 16 + 7) / 8);

    for (int k = 0; k < 3; ++k) {
        int cin = 64 * k, cout = 64 * (k + 1);
        k_zero<<<zblocks, 256, 0, stream>>>(side, zero4);
        k_spmm<<<sblocks, 256, 0, stream>>>(rows, cols, vals, out, cin, side, nnz);
        k_dense<<<dblocks, 256, 0, stream>>>(side, out, out,
                                             W[k][0], W[k][1], W[k][2], W[k][3],
                                             cin, cout);
    }
}